// NMS_75505525064646
// MI455X (gfx1250) — compile-verified
//
#include <hip/hip_runtime.h>
#include <hip/hip_bf16.h>

#define N_IMG   8
#define N_ANCH  25200
#define NCLS    80
#define KPRE    2048
#define MAXDET  300
#define CONF_T  0.25f
#define IOU_T   0.45f
#define MAX_WH  4096.0f

typedef __attribute__((ext_vector_type(2))) float v2f;
typedef __attribute__((ext_vector_type(8))) float v8f;

#if defined(__HIP_DEVICE_COMPILE__) && __has_builtin(__builtin_amdgcn_global_load_async_to_lds_b128)
#define USE_ASYNC 1
typedef int v4i_vs __attribute__((vector_size(16)));
typedef __attribute__((address_space(1))) v4i_vs* g_v4i;   // global (AS1) int4*
typedef __attribute__((address_space(3))) v4i_vs* l_v4i;   // LDS (AS3) int4*
#else
#define USE_ASYNC 0
#endif

// ---------------- workspace layout (bytes) ----------------
#define OFF_KEY   (0)                                   // u32 [8*25200]
#define OFF_XYXY  (OFF_KEY   + N_IMG*N_ANCH*4)          // f32 [8*25200*4]
#define OFF_CLS   (OFF_XYXY  + N_IMG*N_ANCH*16)         // i32 [8*25200]
#define OFF_TSC   (OFF_CLS   + N_IMG*N_ANCH*4)          // f32 [8*2048]
#define OFF_TBOX  (OFF_TSC   + N_IMG*KPRE*4)            // f32 [8*2048*4]
#define OFF_TOFF  (OFF_TBOX  + N_IMG*KPRE*16)           // f32 [8*2048*4]
#define OFF_TCLS  (OFF_TOFF  + N_IMG*KPRE*16)           // i32 [8*2048]
#define OFF_AREA  (OFF_TCLS  + N_IMG*KPRE*4)            // f32 [8*2048]
#define OFF_MASK  (OFF_AREA  + N_IMG*KPRE*4)            // u32 [8*2048*64]
#define MASK_BYTES (N_IMG*KPRE*64*4)

// ---------------- kernel 1: decode + score ----------------
__global__ void nms_decode_kernel(const float* __restrict__ x,
                                  unsigned* __restrict__ key,
                                  float* __restrict__ xyxy,
                                  int* __restrict__ cls) {
    int gid = blockIdx.x * blockDim.x + threadIdx.x;
    if (gid >= N_IMG * N_ANCH) return;
    const float* row = x + (size_t)gid * 85;
    float cx = row[0], cy = row[1], w = row[2], h = row[3], obj = row[4];
    float best = row[5]; int bi = 0;
    #pragma unroll 4
    for (int c = 1; c < NCLS; ++c) {
        float s = row[5 + c];
        if (s > best) { best = s; bi = c; }
    }
    float score = obj * best;          // obj >= 0 => max/argmax commute with obj*
    key[gid] = (score > CONF_T) ? __float_as_uint(score) : 0u;
    xyxy[gid * 4 + 0] = cx - w * 0.5f;
    xyxy[gid * 4 + 1] = cy - h * 0.5f;
    xyxy[gid * 4 + 2] = cx + w * 0.5f;
    xyxy[gid * 4 + 3] = cy + h * 0.5f;
    cls[gid] = bi;
}

// ---------------- bitonic sort (2048 u64 keys, ascending) ----------------
__device__ __forceinline__ void bitonic2048(unsigned long long* k, int tid, int nth) {
    for (unsigned kk = 2; kk <= KPRE; kk <<= 1) {
        for (unsigned j = kk >> 1; j > 0; j >>= 1) {
            __syncthreads();
            for (unsigned e = (unsigned)tid; e < KPRE; e += (unsigned)nth) {
                unsigned p = e ^ j;
                if (p > e) {
                    bool asc = ((e & kk) == 0);
                    unsigned long long a = k[e], b = k[p];
                    if ((a > b) == asc) { k[e] = b; k[p] = a; }
                }
            }
        }
    }
    __syncthreads();
}

// ---------------- kernel 2: radix-select top-2048 + sort + gather ----------------
__global__ void nms_topk_kernel(const unsigned* __restrict__ key,
                                const float* __restrict__ xyxy,
                                const int* __restrict__ cls,
                                float* __restrict__ topScore,
                                float* __restrict__ topBox,
                                float* __restrict__ topOff,
                                int* __restrict__ topCls,
                                float* __restrict__ area) {
    int img = blockIdx.x;
    int tid = threadIdx.x;
    const int NTH = 1024;
    __shared__ unsigned hist[256];
    __shared__ int s_bin, s_cntA, s_cnt, s_tie;
    __shared__ unsigned long long skey[KPRE];

    const unsigned* K = key + (size_t)img * N_ANCH;
    unsigned prefix = 0;
    int needed = KPRE;
    for (int pass = 0; pass < 4; ++pass) {
        int shift = 24 - 8 * pass;
        for (int b = tid; b < 256; b += NTH) hist[b] = 0u;
        __syncthreads();
        for (int e = tid; e < N_ANCH; e += NTH) {
            unsigned u = K[e];
            if (pass == 0 || (u >> (shift + 8)) == prefix)
                atomicAdd(&hist[(u >> shift) & 255u], 1u);
        }
        __syncthreads();
        if (tid == 0) {
            int cum = 0, bin = 0;
            for (int b = 255; b >= 0; --b) {
                int c = (int)hist[b];
                if (cum + c >= needed) { bin = b; break; }
                cum += c;
            }
            s_bin = bin; s_cntA = cum;
        }
        __syncthreads();
        prefix = (prefix << 8) | (unsigned)s_bin;
        needed -= s_cntA;
        __syncthreads();
    }
    unsigned T = prefix;
    if (tid == 0) { s_cnt = 0; s_tie = 0; }
    __syncthreads();
    for (int e = tid; e < N_ANCH; e += NTH) {
        unsigned u = K[e];
        int slot = -1;
        if (u > T) slot = atomicAdd(&s_cnt, 1);
        else if (u == T) {
            int t = atomicAdd(&s_tie, 1);
            if (t < needed) slot = atomicAdd(&s_cnt, 1);
        }
        if (slot >= 0)
            skey[slot] = ~((((unsigned long long)u) << 32) |
                           (unsigned long long)(0xFFFFFFFFu - (unsigned)e));
    }
    __syncthreads();
    bitonic2048(skey, tid, NTH);   // ascending of ~key == descending (score, then idx asc)

    for (int r = tid; r < KPRE; r += NTH) {
        unsigned long long kk = ~skey[r];
        unsigned u = (unsigned)(kk >> 32);
        int e = (int)(0xFFFFFFFFu - (unsigned)kk);
        int o = img * KPRE + r;
        float sc = u ? __uint_as_float(u) : -1.0f;
        topScore[o] = sc;
        float b0 = 0.f, b1 = 0.f, b2 = 0.f, b3 = 0.f; int c = 0;
        if (u) {
            const float* bx = xyxy + ((size_t)img * N_ANCH + e) * 4;
            b0 = bx[0]; b1 = bx[1]; b2 = bx[2]; b3 = bx[3];
            c = cls[(size_t)img * N_ANCH + e];
        }
        topBox[o * 4 + 0] = b0; topBox[o * 4 + 1] = b1;
        topBox[o * 4 + 2] = b2; topBox[o * 4 + 3] = b3;
        float off = (float)c * MAX_WH;
        topOff[o * 4 + 0] = b0 + off; topOff[o * 4 + 1] = b1 + off;
        topOff[o * 4 + 2] = b2 + off; topOff[o * 4 + 3] = b3 + off;
        topCls[o] = c;
        area[o] = (b2 - b0) * (b3 - b1);
    }
}

// ---------------- kernel 3: IoU bitmask ----------------
// One block per (img, rowTile). All 2048 offset boxes + areas of the image are
// staged in LDS once (async-to-LDS on CDNA5), then 8 waves sweep the column
// tiles. The IoU denominator's (a_i + a_j) outer sum is one
// v_wmma_f32_16x16x4_f32 per 16x16 tile.
__global__ void nms_iou_kernel(const float* __restrict__ topOff,
                               const float* __restrict__ area,
                               unsigned* __restrict__ mask) {
    __shared__ float sOff[KPRE * 4];   // 32 KB
    __shared__ float sArea[KPRE];      // 8 KB
    int img = blockIdx.x >> 7;
    int rT  = blockIdx.x & 127;
    int tid = threadIdx.x;

    const float* srcO = topOff + (size_t)img * KPRE * 4;
    const float* srcA = area + (size_t)img * KPRE;

#if USE_ASYNC
    // 16 B per lane per issue; ASYNCcnt tracked, drained before the barrier.
    for (int e = tid * 4; e < KPRE * 4; e += 256 * 4)
        __builtin_amdgcn_global_load_async_to_lds_b128(
            (g_v4i)(srcO + e), (l_v4i)(&sOff[e]), 0, 0);
    for (int e = tid * 4; e < KPRE; e += 256 * 4)
        __builtin_amdgcn_global_load_async_to_lds_b128(
            (g_v4i)(srcA + e), (l_v4i)(&sArea[e]), 0, 0);
#if __has_builtin(__builtin_amdgcn_s_wait_asynccnt)
    __builtin_amdgcn_s_wait_asynccnt(0);
#else
    asm volatile("s_wait_asynccnt 0x0" ::: "memory");
#endif
#else
    for (int e = tid; e < KPRE * 4; e += 256) sOff[e] = srcO[e];
    for (int e = tid; e < KPRE; e += 256) sArea[e] = srcA[e];
#endif
    __syncthreads();

    int lane = tid & 31;
    int waveId = tid >> 5;
    int half = (lane >> 4) & 1;

    // Row-block constants (fixed for the whole block)
    v2f av; av.x = (lane < 16) ? sArea[rT * 16 + lane] : 0.0f;
            av.y = (lane < 16) ? 1.0f : 0.0f;
    float ib[8][4];
    #pragma unroll
    for (int r = 0; r < 8; ++r) {
        int i = rT * 16 + r + half * 8;                    // matches C/D VGPR layout
        ib[r][0] = sOff[i * 4 + 0]; ib[r][1] = sOff[i * 4 + 1];
        ib[r][2] = sOff[i * 4 + 2]; ib[r][3] = sOff[i * 4 + 3];
    }
    unsigned* Mrow = mask + ((size_t)img * KPRE + rT * 16) * 64;

    for (int cT = waveId; cT <= rT; cT += 8) {             // only j < i matters
        // B matrix 4x16: v0 lanes 0-15 = row K0 (ones), lanes 16-31 = row K1 (a_j)
        v2f bv; bv.x = (lane < 16) ? 1.0f : sArea[cT * 16 + (lane - 16)];
                bv.y = 0.0f;
        v8f cz = {0.f, 0.f, 0.f, 0.f, 0.f, 0.f, 0.f, 0.f};
        v8f d = __builtin_amdgcn_wmma_f32_16x16x4_f32(false, av, false, bv,
                                                      (short)0, cz, false, false);
        int j = cT * 16 + (lane & 15);
        float jx1 = sOff[j * 4 + 0], jy1 = sOff[j * 4 + 1];
        float jx2 = sOff[j * 4 + 2], jy2 = sOff[j * 4 + 3];

        unsigned ball[8];
        #pragma unroll
        for (int r = 0; r < 8; ++r) {
            float w = fmaxf(fminf(ib[r][2], jx2) - fmaxf(ib[r][0], jx1), 0.0f);
            float h = fmaxf(fminf(ib[r][3], jy2) - fmaxf(ib[r][1], jy1), 0.0f);
            float inter = w * h;
            float iou = inter / (d[r] - inter + 1e-7f);    // d[r] = a_i + a_j via WMMA
            ball[r] = __builtin_amdgcn_ballot_w32(iou > IOU_T);
        }
        if (lane == 0) {
            int w = cT >> 1;
            int sh = (cT & 1) * 16;
            #pragma unroll
            for (int r = 0; r < 8; ++r) {
                unsigned lo = ball[r] & 0xFFFFu;           // rows rT*16+r
                unsigned hi = ball[r] >> 16;               // rows rT*16+r+8
                if (lo) atomicOr(&Mrow[(size_t)r * 64 + w], lo << sh);
                if (hi) atomicOr(&Mrow[(size_t)(r + 8) * 64 + w], hi << sh);
            }
        }
    }
}

// ---------------- kernel 4: serial greedy scan + top-300 + output ----------------
__global__ void nms_scan_kernel(const unsigned* __restrict__ mask,
                                const float* __restrict__ topScore,
                                const float* __restrict__ topBox,
                                const int* __restrict__ topCls,
                                float* __restrict__ out) {
    int img = blockIdx.x;
    int tid = threadIdx.x;
    const int NTH = 1024;
    __shared__ unsigned keepW[64];
    __shared__ unsigned long long skey[KPRE];

    if (tid < 64) keepW[tid] = 0u;
    __syncthreads();

    if (tid < 32) {   // one wave runs the serial greedy scan; lane owns words {tid, tid+32}
        volatile unsigned* kw = keepW;
        const unsigned* M = mask + (size_t)img * KPRE * 64;
        const float* S = topScore + (size_t)img * KPRE;
        for (int i = 0; i < KPRE; ++i) {
            int iw = i >> 5, ib = i & 31;
            unsigned m0 = M[(size_t)i * 64 + tid];
            unsigned m1 = M[(size_t)i * 64 + tid + 32];
            unsigned c0 = (tid < iw) ? 0xFFFFFFFFu : ((tid == iw) ? ((1u << ib) - 1u) : 0u);
            int w1 = tid + 32;
            unsigned c1 = (w1 < iw) ? 0xFFFFFFFFu : ((w1 == iw) ? ((1u << ib) - 1u) : 0u);
            unsigned s = (kw[tid] & m0 & c0) | (kw[w1] & m1 & c1);
            unsigned anyb = __builtin_amdgcn_ballot_w32(s != 0u);
            bool valid = S[i] > 0.0f;
            if (anyb == 0u && valid) {
                if (tid == (iw & 31)) kw[iw] |= (1u << ib);
            }
        }
    }
    __syncthreads();

    for (int i = tid; i < KPRE; i += NTH) {
        int iw = i >> 5, ib = i & 31;
        bool keep = (keepW[iw] >> ib) & 1u;
        float sc = topScore[(size_t)img * KPRE + i];
        unsigned u = keep ? __float_as_uint(sc) : 0u;      // kept scores are > 0
        skey[i] = ~((((unsigned long long)u) << 32) |
                    (unsigned long long)(0xFFFFFFFFu - (unsigned)i));
    }
    __syncthreads();
    bitonic2048(skey, tid, NTH);

    for (int r = tid; r < MAXDET; r += NTH) {
        unsigned long long kk = ~skey[r];
        unsigned u = (unsigned)(kk >> 32);
        int i = (int)(0xFFFFFFFFu - (unsigned)kk);
        float* o = out + ((size_t)img * MAXDET + r) * 6;
        if (u) {
            const float* b = topBox + ((size_t)img * KPRE + i) * 4;
            o[0] = b[0]; o[1] = b[1]; o[2] = b[2]; o[3] = b[3];
            o[4] = __uint_as_float(u);
            o[5] = (float)topCls[(size_t)img * KPRE + i];
        } else {
            o[0] = 0.f; o[1] = 0.f; o[2] = 0.f; o[3] = 0.f; o[4] = 0.f; o[5] = 0.f;
        }
    }
}

extern "C" void kernel_launch(void* const* d_in, const int* in_sizes, int n_in,
                              void* d_out, int out_size, void* d_ws, size_t ws_size,
                              hipStream_t stream) {
    const float* x = (const float*)d_in[0];
    float* out = (float*)d_out;
    char* ws = (char*)d_ws;

    unsigned* key      = (unsigned*)(ws + OFF_KEY);
    float*    xyxy     = (float*)   (ws + OFF_XYXY);
    int*      cls      = (int*)     (ws + OFF_CLS);
    float*    topScore = (float*)   (ws + OFF_TSC);
    float*    topBox   = (float*)   (ws + OFF_TBOX);
    float*    topOff   = (float*)   (ws + OFF_TOFF);
    int*      topCls   = (int*)     (ws + OFF_TCLS);
    float*    area     = (float*)   (ws + OFF_AREA);
    unsigned* mask     = (unsigned*)(ws + OFF_MASK);

    (void)hipMemsetAsync(mask, 0, MASK_BYTES, stream);   // atomicOr target must start at 0

    int total = N_IMG * N_ANCH;
    nms_decode_kernel<<<(total + 255) / 256, 256, 0, stream>>>(x, key, xyxy, cls);
    nms_topk_kernel<<<N_IMG, 1024, 0, stream>>>(key, xyxy, cls, topScore, topBox,
                                                topOff, topCls, area);
    // one block per (img, rowTile): 8 * 128 blocks, 8 waves each
    nms_iou_kernel<<<N_IMG * 128, 256, 0, stream>>>(topOff, area, mask);
    nms_scan_kernel<<<N_IMG, 1024, 0, stream>>>(mask, topScore, topBox, topCls, out);
}